// MixHopConv_936302871064
// MI455X (gfx1250) — compile-verified
//
#include <hip/hip_runtime.h>
#include <hip/hip_bf16.h>

typedef __attribute__((ext_vector_type(2))) float v2f;
typedef __attribute__((ext_vector_type(8))) float v8f;

// ---------------------------------------------------------------------------
// Zero-fill workspace (hop1 | hop2).
// ---------------------------------------------------------------------------
__global__ void mixhop_zero(float* __restrict__ p, long long n) {
  long long i = (long long)blockIdx.x * blockDim.x + threadIdx.x;
  if (i < n) p[i] = 0.0f;
}

// ---------------------------------------------------------------------------
// One propagation hop: dst_feat[dst[e]] += src_feat[src[e]] for all edges.
// 16 threads per edge, each owning a float4 channel chunk (64 channels).
// Feature table (12.8 MB) is L2-resident on MI455X (192 MB L2), so the
// gathers + global_atomic_add_f32 stay on-chip.
// ---------------------------------------------------------------------------
__global__ void mixhop_scatter(const float* __restrict__ src_feat,
                               float* __restrict__ dst_feat,
                               const int* __restrict__ src_idx,
                               const int* __restrict__ dst_idx,
                               int n_edges) {
  long long t = (long long)blockIdx.x * blockDim.x + threadIdx.x;
  int e  = (int)(t >> 4);
  int c4 = ((int)t & 15) * 4;
  if (e >= n_edges) return;
  int s = src_idx[e];
  int d = dst_idx[e];
  const float4 v = *(const float4*)(src_feat + (long long)s * 64 + c4);
  float* p = dst_feat + (long long)d * 64 + c4;
  atomicAdd(p + 0, v.x);
  atomicAdd(p + 1, v.y);
  atomicAdd(p + 2, v.z);
  atomicAdd(p + 3, v.w);
}

// ---------------------------------------------------------------------------
// out[m][n] = sum_k feats[m][k] * W[n][k] + b[n],  feats = [x | h1 | h2].
// One wave (32 threads) per 16-node strip; 8 n-tiles of 16 -> full 128 cols.
// Uses V_WMMA_F32_16X16X4_F32 (exact fp32) with the ISA 16x16x4 layouts:
//   A: lane l -> row m=l&15, K = 2*(l>>4)+{0,1}           (contiguous v2f)
//   B: lane l -> col n=l&15, K = 2*(l>>4)+{0,1} = W[n][k] (contiguous v2f)
//   D: vgpr v, half h -> row v+8h, col l&15
// 50000 nodes = 3125 full tiles, so EXEC is all-ones (WMMA requirement).
// ---------------------------------------------------------------------------
__global__ __launch_bounds__(32) void mixhop_wmma_gemm(
    const float* __restrict__ x,
    const float* __restrict__ h1,
    const float* __restrict__ h2,
    const float* __restrict__ W,      // [128][192] row-major
    const float* __restrict__ bias,   // [128]
    float* __restrict__ out) {        // [n_nodes][128]
  const int row0 = blockIdx.x * 16;
  const int lane = threadIdx.x;
  const int mn   = lane & 15;   // A-row / B-col within tile
  const int hh   = lane >> 4;   // lane half
  const int m    = row0 + mn;

  v8f acc[8] = {};

  const float* segs[3] = {x, h1, h2};
#pragma unroll
  for (int s = 0; s < 3; ++s) {
    const float* aseg = segs[s] + (long long)m * 64 + 2 * hh;
    const float* wseg = W + 64 * s + 2 * hh;
#pragma unroll
    for (int kk = 0; kk < 64; kk += 4) {
      v2f a = *(const v2f*)(aseg + kk);
#pragma unroll
      for (int j = 0; j < 8; ++j) {
        const int n = j * 16 + mn;
        v2f bv = *(const v2f*)(wseg + (long long)n * 192 + kk);
        acc[j] = __builtin_amdgcn_wmma_f32_16x16x4_f32(
            /*neg_a=*/false, a, /*neg_b=*/false, bv,
            /*c_mod=*/(short)0, acc[j],
            /*reuse_a=*/false, /*reuse_b=*/false);
      }
    }
  }

#pragma unroll
  for (int j = 0; j < 8; ++j) {
    const int n = j * 16 + mn;
    const float bb = bias[n];
#pragma unroll
    for (int v = 0; v < 8; ++v) {
      const int rm = row0 + v + 8 * hh;
      out[(long long)rm * 128 + n] = acc[j][v] + bb;
    }
  }
}

// ---------------------------------------------------------------------------
// Launch: zero ws -> hop1 scatter -> hop2 scatter -> WMMA GEMM.
// ---------------------------------------------------------------------------
extern "C" void kernel_launch(void* const* d_in, const int* in_sizes, int n_in,
                              void* d_out, int out_size, void* d_ws, size_t ws_size,
                              hipStream_t stream) {
  const float* x    = (const float*)d_in[0];
  const int*   ei   = (const int*)d_in[1];
  const float* W    = (const float*)d_in[2];
  const float* bias = (const float*)d_in[3];
  float*       out  = (float*)d_out;

  const int n_nodes = in_sizes[0] / 64;   // 50000
  const int n_edges = in_sizes[1] / 2;    // 800000

  float* h1 = (float*)d_ws;
  float* h2 = h1 + (long long)n_nodes * 64;

  const int* src = ei;
  const int* dst = ei + n_edges;

  // 1) zero hop buffers (contiguous in workspace)
  {
    long long n = 2LL * n_nodes * 64;
    int blocks = (int)((n + 255) / 256);
    mixhop_zero<<<blocks, 256, 0, stream>>>(h1, n);
  }
  // 2) hop1 = scatter_add(x[src] -> dst)
  {
    long long t = (long long)n_edges * 16;
    int blocks = (int)((t + 255) / 256);
    mixhop_scatter<<<blocks, 256, 0, stream>>>(x, h1, src, dst, n_edges);
  }
  // 3) hop2 = scatter_add(h1[src] -> dst)
  {
    long long t = (long long)n_edges * 16;
    int blocks = (int)((t + 255) / 256);
    mixhop_scatter<<<blocks, 256, 0, stream>>>(h1, h2, src, dst, n_edges);
  }
  // 4) out = [x|h1|h2] @ W^T + b  (WMMA f32 16x16x4)
  {
    int tiles = n_nodes / 16;  // 3125 full tiles
    mixhop_wmma_gemm<<<tiles, 32, 0, stream>>>(x, h1, h2, W, bias, out);
  }
}